// UCR_78615081386430
// MI455X (gfx1250) — compile-verified
//
#include <hip/hip_runtime.h>
#include <hip/hip_bf16.h>
#include <cstdint>

// ---------------------------------------------------------------------------
// Problem constants (from the reference)
// ---------------------------------------------------------------------------
#define NU0 20000
#define NI0 15000
#define NU1 4000
#define NI1 3000
#define NN0 (NU0 + NI0)     // 35000
#define NN1 (NU1 + NI1)     // 7000
#define E0  1000000
#define E1  200000
#define DD  64
#define LL  3

typedef __attribute__((ext_vector_type(2))) float v2f;
typedef __attribute__((ext_vector_type(8))) float v8f;

// D = A(16x4) * B(4x16) + C, fp32 WMMA (CDNA5)
__device__ __forceinline__ v8f wmma4(v2f a, v2f b, v8f c) {
    return __builtin_amdgcn_wmma_f32_16x16x4_f32(
        /*neg_a=*/false, a, /*neg_b=*/false, b,
        /*c_mod=*/(short)0, c, /*reuse_a=*/false, /*reuse_b=*/false);
}

__device__ __forceinline__ float leaky01(float x) {
    return x > 0.0f ? x : 0.01f * x;
}

// CDNA5 async copy: global -> LDS, 16 bytes per lane, tracked by ASYNCcnt.
// lds_dst is a generic pointer to __shared__; its low 32 bits are the
// wave-relative LDS byte offset (ISA: LDS aperture addr[31:0]).
__device__ __forceinline__ void async_ld_b128(void* lds_dst, const void* g_src) {
    uint32_t lds_off = (uint32_t)(uintptr_t)lds_dst;
    asm volatile("global_load_async_to_lds_b128 %0, %1, off"
                 :: "v"(lds_off), "v"(g_src)
                 : "memory");
}

__device__ __forceinline__ void wait_async0() {
    asm volatile("s_wait_asynccnt 0x0" ::: "memory");
}

// ---------------------------------------------------------------------------
// ego = concat(ue, ie); also writes cols [0,64) of all_e (outs[0], unnormalized)
// ---------------------------------------------------------------------------
__global__ void build_ego_kernel(const float* __restrict__ ue,
                                 const float* __restrict__ ie,
                                 int nu, int n,
                                 float* __restrict__ ego,
                                 float* __restrict__ all_e) {
    long i = (long)blockIdx.x * blockDim.x + threadIdx.x;
    long total = (long)n * DD;
    if (i >= total) return;
    int row = (int)(i >> 6);
    int col = (int)(i & 63);
    float v = (row < nu) ? ue[i] : ie[i - (long)nu * DD];
    ego[i] = v;
    all_e[(size_t)row * 256 + col] = v;
}

__global__ void zero_kernel(float* __restrict__ p, long n) {
    long i = (long)blockIdx.x * blockDim.x + threadIdx.x;
    if (i < n) p[i] = 0.0f;
}

// ---------------------------------------------------------------------------
// side[rows[e]] += vals[e] * ego[cols[e]]   (16 threads per edge, 4 floats each)
// ---------------------------------------------------------------------------
__global__ void spmm_scatter_kernel(const float* __restrict__ vals,
                                    const int* __restrict__ rows,
                                    const int* __restrict__ cols,
                                    const float* __restrict__ ego,
                                    float* __restrict__ side, int E) {
    long t = (long)blockIdx.x * blockDim.x + threadIdx.x;
    int e = (int)(t >> 4);
    if (e >= E) return;
    int q = ((int)t & 15) * 4;
    float v = vals[e];
    const float4 x = *(const float4*)(ego + (size_t)cols[e] * DD + q);
    float* dst = side + (size_t)rows[e] * DD + q;
    atomicAdd(dst + 0, v * x.x);
    atomicAdd(dst + 1, v * x.y);
    atomicAdd(dst + 2, v * x.z);
    atomicAdd(dst + 3, v * x.w);
}

// ---------------------------------------------------------------------------
// One NGCF layer:
//   sum_e = leaky(side @ gcW + gcb);  bi_e = leaky((ego*side) @ biW + bib)
//   ego_next = sum_e + bi_e;  all_e[:, out_col:out_col+64] = normalize(ego_next)
// One wave per 16-row tile; two fused 16x64x64 WMMA GEMMs.
// ---------------------------------------------------------------------------
__global__ void __launch_bounds__(128)
ngcf_layer_kernel(const float* __restrict__ side,
                  const float* __restrict__ ego,
                  const float* __restrict__ gcW, const float* __restrict__ gcb,
                  const float* __restrict__ biW, const float* __restrict__ bib,
                  float* __restrict__ ego_next,
                  float* __restrict__ all_e,
                  int n, int out_col) {
    const int lane = threadIdx.x & 31;
    const int wave = threadIdx.x >> 5;
    const int tile = blockIdx.x * 4 + wave;   // 16-row tile index
    const int row0 = tile * 16;
    if (row0 >= n) return;                    // wave-uniform early out

    const int half = lane >> 4;               // 0 or 1
    const int l16  = lane & 15;
    const int koff = half * 2;
    int ar = row0 + l16;                      // A-fragment row (clamped for tail)
    if (ar >= n) ar = n - 1;
    const float* srow = side + (size_t)ar * DD;
    const float* erow = ego  + (size_t)ar * DD;

    v8f accS[4], accB[4];
#pragma unroll
    for (int t = 0; t < 4; ++t)
#pragma unroll
        for (int i = 0; i < 8; ++i) { accS[t][i] = 0.0f; accB[t][i] = 0.0f; }

#pragma unroll
    for (int k = 0; k < 16; ++k) {
        const int kk = 4 * k + koff;
        v2f aS, aB;
        float s0 = srow[kk], s1 = srow[kk + 1];
        aS.x = s0;             aS.y = s1;
        aB.x = s0 * erow[kk];  aB.y = s1 * erow[kk + 1];
#pragma unroll
        for (int t = 0; t < 4; ++t) {
            const int nb = 16 * t + l16;
            v2f bS, bB;
            bS.x = gcW[kk * DD + nb]; bS.y = gcW[(kk + 1) * DD + nb];
            bB.x = biW[kk * DD + nb]; bB.y = biW[(kk + 1) * DD + nb];
            accS[t] = wmma4(aS, bS, accS[t]);
            accB[t] = wmma4(aB, bB, accB[t]);
        }
    }

    // Epilogue: bias + leaky, sum, squared row sums.
    float vout[4][8];
    float rss[8];
#pragma unroll
    for (int r = 0; r < 8; ++r) rss[r] = 0.0f;
#pragma unroll
    for (int t = 0; t < 4; ++t) {
        const float bS = gcb[16 * t + l16];
        const float bB = bib[16 * t + l16];
#pragma unroll
        for (int r = 0; r < 8; ++r) {
            float x = leaky01(accS[t][r] + bS);
            float y = leaky01(accB[t][r] + bB);
            float e = x + y;
            vout[t][r] = e;
            rss[r] += e * e;
        }
    }
    // Row-norm: row (r + 8*half) lives entirely in this 16-lane half.
    float inv[8];
#pragma unroll
    for (int r = 0; r < 8; ++r) {
        float s = rss[r];
        s += __shfl_xor(s, 1, 16);
        s += __shfl_xor(s, 2, 16);
        s += __shfl_xor(s, 4, 16);
        s += __shfl_xor(s, 8, 16);
        float nrm = sqrtf(s);
        inv[r] = 1.0f / fmaxf(nrm, 1e-12f);
    }
#pragma unroll
    for (int t = 0; t < 4; ++t) {
#pragma unroll
        for (int r = 0; r < 8; ++r) {
            const int row = row0 + r + half * 8;
            if (row < n) {
                const int col = 16 * t + l16;
                ego_next[(size_t)row * DD + col] = vout[t][r];
                all_e[(size_t)row * 256 + out_col + col] = vout[t][r] * inv[r];
            }
        }
    }
}

// ---------------------------------------------------------------------------
// out[M,256] = C0[M,256] + A[M,K] @ B[K,256]
// fp32 WMMA with double-buffered ASYNC global->LDS staging (gfx1250 TDM-class
// path, ASYNCcnt). Block: 128 threads (4 waves).
//   blockIdx.x -> 64-col tile, blockIdx.y -> 64-row tile, K-chunk = 32.
// LDS strides (36 / 68 dwords) are multiples of 4 dwords => every async b128
// destination is 16B-aligned, while A-fragment reads still spread banks.
// ---------------------------------------------------------------------------
#define TK 32

__global__ void __launch_bounds__(128)
map_gemm_kernel(const float* __restrict__ A,
                const float* __restrict__ B,
                const float* __restrict__ C0,
                float* __restrict__ out,
                int M, int K) {
    __shared__ float As[2][64][TK + 4];   // 64 rows x 32 k, stride 144B
    __shared__ float Bs[2][TK][68];       // 32 k  x 64 n, stride 272B

    const int tid   = threadIdx.x;
    const int lane  = tid & 31;
    const int wave  = tid >> 5;
    const int half  = lane >> 4;
    const int l16   = lane & 15;
    const int koff  = half * 2;
    const int nbase = blockIdx.x * 64;
    const int mbase = blockIdx.y * 64;

    v8f acc[4];
#pragma unroll
    for (int t = 0; t < 4; ++t)
#pragma unroll
        for (int i = 0; i < 8; ++i) acc[t][i] = 0.0f;

    const int nchunks = (K + TK - 1) / TK;

    // Issue one chunk of async b128 copies (8 per thread). Addresses are
    // clamped at the M/K edges (EXEC stays full; no OOB faults); the K-tail
    // garbage is zeroed in LDS by the fixup below before it is consumed.
    auto issue = [&](int buf, int k0) {
#pragma unroll
        for (int i = 0; i < 4; ++i) {           // A: 64 x 32 = 512 segs
            const int s   = tid + i * 128;
            const int row = s >> 3;             // 0..63
            const int col = (s & 7) * 4;        // 0..28
            int grow = mbase + row; if (grow >= M) grow = M - 1;
            int gcol = k0 + col;    if (gcol > K - 4) gcol = K - 4;
            async_ld_b128(&As[buf][row][col], A + (size_t)grow * K + gcol);
        }
#pragma unroll
        for (int i = 0; i < 4; ++i) {           // B: 32 x 64 = 512 segs
            const int s   = tid + i * 128;
            const int row = s >> 4;             // 0..31
            const int col = (s & 15) * 4;       // 0..60
            int kb = k0 + row; if (kb > K - 1) kb = K - 1;
            async_ld_b128(&Bs[buf][row][col], B + (size_t)kb * 256 + nbase + col);
        }
    };

    issue(0, 0);

    for (int c = 0; c < nchunks; ++c) {
        const int buf = c & 1;
        const int k0  = c * TK;

        wait_async0();        // this wave's chunk-c copies have landed in LDS
        __syncthreads();      // everyone's have

        if (K - k0 < TK) {    // block-uniform K-tail fixup
            const int kend = K - k0;
            for (int idx = tid; idx < 64 * (TK - kend); idx += 128) {
                const int row = idx / (TK - kend);
                const int col = kend + idx % (TK - kend);
                As[buf][row][col] = 0.0f;
            }
            for (int idx = tid; idx < (TK - kend) * 64; idx += 128) {
                const int row = kend + idx / 64;
                const int col = idx % 64;
                Bs[buf][row][col] = 0.0f;
            }
            __syncthreads();
        }

        if (c + 1 < nchunks) issue(buf ^ 1, k0 + TK);  // overlap with compute

        const int arow = wave * 16 + l16;
#pragma unroll
        for (int ks = 0; ks < TK / 4; ++ks) {
            const int kk = 4 * ks + koff;
            v2f a;
            a.x = As[buf][arow][kk];
            a.y = As[buf][arow][kk + 1];
#pragma unroll
            for (int t = 0; t < 4; ++t) {
                v2f b;
                b.x = Bs[buf][kk][16 * t + l16];
                b.y = Bs[buf][kk + 1][16 * t + l16];
                acc[t] = wmma4(a, b, acc[t]);
            }
        }
    }

#pragma unroll
    for (int t = 0; t < 4; ++t) {
#pragma unroll
        for (int rr = 0; rr < 8; ++rr) {
            const int row = mbase + wave * 16 + rr + half * 8;
            if (row < M) {
                const int col = nbase + 16 * t + l16;
                const size_t o = (size_t)row * 256 + col;
                out[o] = acc[t][rr] + C0[o];
            }
        }
    }
}

// ---------------------------------------------------------------------------
// Host side
// ---------------------------------------------------------------------------
static void run_graph(const float* ue, const float* ie,
                      const float* gcW, const float* gcb,
                      const float* biW, const float* bib,
                      const float* vals, const int* rows, const int* cols,
                      int nu, int n, int E,
                      float* egoA, float* egoB, float* side, float* alle,
                      hipStream_t stream) {
    const long ne = (long)n * DD;
    build_ego_kernel<<<(unsigned)((ne + 255) / 256), 256, 0, stream>>>(
        ue, ie, nu, n, egoA, alle);
    float* cur = egoA;
    float* nxt = egoB;
    for (int l = 0; l < LL; ++l) {
        zero_kernel<<<(unsigned)((ne + 255) / 256), 256, 0, stream>>>(side, ne);
        const long nt = (long)E * 16;
        spmm_scatter_kernel<<<(unsigned)((nt + 255) / 256), 256, 0, stream>>>(
            vals, rows, cols, cur, side, E);
        const int ntiles = (n + 15) / 16;
        ngcf_layer_kernel<<<(unsigned)((ntiles + 3) / 4), 128, 0, stream>>>(
            side, cur, gcW + l * DD * DD, gcb + l * DD,
            biW + l * DD * DD, bib + l * DD,
            nxt, alle, n, DD * (l + 1));
        float* tmp = cur; cur = nxt; nxt = tmp;
    }
}

extern "C" void kernel_launch(void* const* d_in, const int* in_sizes, int n_in,
                              void* d_out, int out_size, void* d_ws, size_t ws_size,
                              hipStream_t stream) {
    const float* ue0   = (const float*)d_in[0];
    const float* ie0   = (const float*)d_in[1];
    const float* gcW0  = (const float*)d_in[2];
    const float* gcb0  = (const float*)d_in[3];
    const float* biW0  = (const float*)d_in[4];
    const float* bib0  = (const float*)d_in[5];
    const float* ue1   = (const float*)d_in[6];
    const float* ie1   = (const float*)d_in[7];
    const float* gcW1  = (const float*)d_in[8];
    const float* gcb1  = (const float*)d_in[9];
    const float* biW1  = (const float*)d_in[10];
    const float* bib1  = (const float*)d_in[11];
    const float* mapu  = (const float*)d_in[12];
    const float* mapi  = (const float*)d_in[13];
    const float* vals0 = (const float*)d_in[14];
    const float* vals1 = (const float*)d_in[15];
    const int*   rows0 = (const int*)d_in[16];
    const int*   cols0 = (const int*)d_in[17];
    const int*   rows1 = (const int*)d_in[18];
    const int*   cols1 = (const int*)d_in[19];

    float* ws = (float*)d_ws;
    size_t off = 0;
    float* ego0a = ws + off; off += (size_t)NN0 * DD;
    float* ego0b = ws + off; off += (size_t)NN0 * DD;
    float* side0 = ws + off; off += (size_t)NN0 * DD;
    float* alle0 = ws + off; off += (size_t)NN0 * 256;
    float* ego1a = ws + off; off += (size_t)NN1 * DD;
    float* ego1b = ws + off; off += (size_t)NN1 * DD;
    float* side1 = ws + off; off += (size_t)NN1 * DD;
    float* alle1 = ws + off; off += (size_t)NN1 * 256;
    (void)ws_size; (void)in_sizes; (void)n_in; (void)out_size;

    // Tower 0 (big graph) and tower 1 (small graph)
    run_graph(ue0, ie0, gcW0, gcb0, biW0, bib0, vals0, rows0, cols0,
              NU0, NN0, E0, ego0a, ego0b, side0, alle0, stream);
    run_graph(ue1, ie1, gcW1, gcb1, biW1, bib1, vals1, rows1, cols1,
              NU1, NN1, E1, ego1a, ego1b, side1, alle1, stream);

    float* out = (float*)d_out;
    // user_embd = ug0 + mapu @ ug1  : M=20000, K=4000
    map_gemm_kernel<<<dim3(4, (NU0 + 63) / 64), 128, 0, stream>>>(
        mapu, alle1, alle0, out, NU0, NU1);
    // item_embd = ig0 + mapi @ ig1  : M=15000, K=3000
    map_gemm_kernel<<<dim3(4, (NI0 + 63) / 64), 128, 0, stream>>>(
        mapi, alle1 + (size_t)NU1 * 256, alle0 + (size_t)NU0 * 256,
        out + (size_t)NU0 * 256, NI0, NI1);
}